// Speller_1090921693204
// MI455X (gfx1250) — compile-verified
//
#include <hip/hip_runtime.h>
#include <math.h>

// Problem sizes (fixed by the reference)
#define B_   512
#define T_   2048
#define H_   128
#define E_   256
#define V_   34
#define G4H  512   // 4*H

typedef float v2f __attribute__((ext_vector_type(2)));
typedef float v8f __attribute__((ext_vector_type(8)));

// ---------------------------------------------------------------------------
// CDNA5 async global->LDS copy (b128 = 16B/lane) with graceful fallback
// ---------------------------------------------------------------------------
#if defined(__has_builtin)
#  if __has_builtin(__builtin_amdgcn_global_load_async_to_lds_b128) && \
      __has_builtin(__builtin_amdgcn_s_wait_asynccnt)
#    define HAVE_ASYNC_LDS 1
#  endif
#endif
#ifndef HAVE_ASYNC_LDS
#  define HAVE_ASYNC_LDS 0
#endif

typedef int v4i_vs __attribute__((vector_size(16)));
typedef __attribute__((address_space(1))) v4i_vs* gptr16;   // global 16B chunk
typedef __attribute__((address_space(3))) v4i_vs* lptr16;   // LDS 16B chunk

__device__ __forceinline__ void async_copy16(const float* g, float* l) {
#if HAVE_ASYNC_LDS
  __builtin_amdgcn_global_load_async_to_lds_b128(
      (gptr16)(g), (lptr16)(l), /*offset=*/0, /*cpol=*/0);
#else
  *(float4*)l = *(const float4*)g;   // synchronous fallback (barrier covers it)
#endif
}

#if HAVE_ASYNC_LDS
#  define WAIT_ASYNC(n) __builtin_amdgcn_s_wait_asynccnt(n)
#else
#  define WAIT_ASYNC(n) ((void)0)
#endif

// ---------------------------------------------------------------------------
// Column-concat packers (build row-major [rows, w0+w1(+w2)] panels in ws)
// ---------------------------------------------------------------------------
__global__ void pack2_kernel(const float* __restrict__ s0, int w0,
                             const float* __restrict__ s1, int w1,
                             float* __restrict__ dst, int rows) {
  int K = w0 + w1;
  int idx = blockIdx.x * blockDim.x + threadIdx.x;
  if (idx >= rows * K) return;
  int r = idx / K, k = idx - r * K;
  dst[idx] = (k < w0) ? s0[r * w0 + k] : s1[r * w1 + (k - w0)];
}

__global__ void pack3_kernel(const float* __restrict__ s0, int w0,
                             const float* __restrict__ s1, int w1,
                             const float* __restrict__ s2, int w2,
                             float* __restrict__ dst, int rows) {
  int K = w0 + w1 + w2;
  int idx = blockIdx.x * blockDim.x + threadIdx.x;
  if (idx >= rows * K) return;
  int r = idx / K, k = idx - r * K;
  float v;
  if (k < w0)            v = s0[r * w0 + k];
  else if (k < w0 + w1)  v = s1[r * w1 + (k - w0)];
  else                   v = s2[r * w2 + (k - w0 - w1)];
  dst[idx] = v;
}

// ---------------------------------------------------------------------------
// FP32 WMMA NT-GEMM: C[M,N] = A[M,K] * Bw[N,K]^T  (all row-major)
// One wave per 16x16 C tile; V_WMMA_F32_16X16X4_F32, K stepped by 4.
// ---------------------------------------------------------------------------
__global__ void wmma_gemm_nt(const float* __restrict__ A,
                             const float* __restrict__ Bw,
                             float* __restrict__ C,
                             int M, int N, int K) {
  int ntiles = (M >> 4) * (N >> 4);
  int tile = blockIdx.x * (blockDim.x >> 5) + (threadIdx.x >> 5);
  if (tile >= ntiles) return;                 // wave-uniform: EXEC stays all-1s
  int tn = tile & ((N >> 4) - 1);             // N is a power of two here
  int tm = tile / (N >> 4);
  int lane = threadIdx.x & 31;
  int half = lane >> 4;                       // khi selector
  int mn   = lane & 15;                       // row (A) / col (B) within tile

  const float* aRow = A  + (size_t)(tm * 16 + mn) * K + 2 * half;
  const float* bRow = Bw + (size_t)(tn * 16 + mn) * K + 2 * half;

  v8f acc = {};
  for (int k = 0; k < K; k += 4) {
    v2f a = *(const v2f*)(aRow + k);
    v2f b = *(const v2f*)(bRow + k);
    acc = __builtin_amdgcn_wmma_f32_16x16x4_f32(
        false, a, false, b, (short)0, acc, false, false);
  }
  int row = tm * 16 + half * 8;
  int col = tn * 16 + mn;
#pragma unroll
  for (int r = 0; r < 8; ++r)
    C[(size_t)(row + r) * N + col] = acc[r];
}

// ---------------------------------------------------------------------------
// LSTM pointwise: gates G[b, 0:4H] (pre-bias, order i,f,g,o) -> h_new, c_new
// ---------------------------------------------------------------------------
__global__ void lstm_elem(const float* __restrict__ G,
                          const float* __restrict__ bih,
                          const float* __restrict__ bhh,
                          const float* __restrict__ cprev,
                          float* __restrict__ hout,
                          float* __restrict__ cout) {
  int idx = blockIdx.x * blockDim.x + threadIdx.x;
  if (idx >= B_ * H_) return;
  int b = idx / H_, h = idx - b * H_;
  const float* g = G + (size_t)b * G4H;
  float gi = g[h]          + bih[h]          + bhh[h];
  float gf = g[H_ + h]     + bih[H_ + h]     + bhh[H_ + h];
  float gg = g[2 * H_ + h] + bih[2 * H_ + h] + bhh[2 * H_ + h];
  float go = g[3 * H_ + h] + bih[3 * H_ + h] + bhh[3 * H_ + h];
  float si = 1.f / (1.f + __expf(-gi));
  float sf = 1.f / (1.f + __expf(-gf));
  float so = 1.f / (1.f + __expf(-go));
  float cn = sf * cprev[idx] + si * tanhf(gg);
  cout[idx] = cn;
  hout[idx] = so * tanhf(cn);
}

// ---------------------------------------------------------------------------
// Fused attention: one workgroup (8 waves, 256 threads) per batch row b.
// Phase 1: hk rows staged to LDS via async b128 DMA (double-buffered, XOR
//          bank swizzle); each thread owns a (t, 16-float h-segment) slice,
//          8-way LDS partial reduce -> scores. No cross-lane shuffles.
// Phase 2: block softmax * mask, renormalized (exact fusion).
// Phase 3: c[b,:] from hv with direct coalesced b128 loads, 4x unrolled.
// Streams hk[b] and hv[b] exactly once each -> HBM-bandwidth bound.
// ---------------------------------------------------------------------------
#define CHUNK 32   // t-rows per staged chunk (32 rows * 512B = 16KB per buffer)

__global__ __launch_bounds__(256) void attention_kernel(
    const float* __restrict__ hk,
    const float* __restrict__ hv,
    const float* __restrict__ sh1,
    const float* __restrict__ mask,
    float* __restrict__ c_out,
    float* __restrict__ att_out) {
  __shared__ float buf[2][CHUNK * H_];   // 2 x 16KB, XOR-swizzled rows
  __shared__ float sc[T_];               // 8KB scores/attn
  __shared__ float pr[256];              // partial dot products
  __shared__ float red[8];
  __shared__ float cacc[8 * H_];

  int b = blockIdx.x;
  int tid = threadIdx.x;                 // 0..255
  int wave = tid >> 5, lane = tid & 31;
  size_t base = (size_t)b * T_ * H_;

  // ---- phase 1: scores via async-staged LDS tiles ----
  // thread -> t-row tl = tid&31, h-segment seg = tid>>5 (16 floats = 4 units)
  int tl = tid & 31, seg = tid >> 5;
  float4 qv[4];
#pragma unroll
  for (int j = 0; j < 4; ++j)
    qv[j] = *(const float4*)(sh1 + (size_t)b * H_ + seg * 16 + 4 * j);

  // wave w DMAs rows r = w*4..w*4+3 of each chunk; lane l moves 16B unit l,
  // landing at swizzled unit (l ^ (r&7)) of the LDS row.
  auto issue_chunk = [&](int ch, int pb) {
#pragma unroll
    for (int j = 0; j < 4; ++j) {
      int r = wave * 4 + j;
      const float* g = hk + base + (size_t)(ch * CHUNK + r) * H_ + lane * 4;
      float* l = &buf[pb][r * H_ + ((lane ^ (r & 7)) << 2)];
      async_copy16(g, l);
    }
  };

  const int NCH = T_ / CHUNK;            // 64 chunks
  issue_chunk(0, 0);
  for (int ch = 0; ch < NCH; ++ch) {
    int pb = ch & 1;
    if (ch + 1 < NCH) {
      issue_chunk(ch + 1, (ch + 1) & 1);
      WAIT_ASYNC(4);                     // chunk ch resident, ch+1 in flight
    } else {
      WAIT_ASYNC(0);
    }
    __syncthreads();

    float s = 0.f;
#pragma unroll
    for (int j = 0; j < 4; ++j) {
      int u = seg * 4 + j;               // 16B unit within the row
      float4 k4 = *(const float4*)&buf[pb][tl * H_ + ((u ^ (tl & 7)) << 2)];
      float4 qj = qv[j];
      s += k4.x * qj.x + k4.y * qj.y + k4.z * qj.z + k4.w * qj.w;
    }
    pr[tid] = s;
    __syncthreads();
    if (tid < CHUNK) {
      float t8 = 0.f;
#pragma unroll
      for (int g8 = 0; g8 < 8; ++g8) t8 += pr[g8 * 32 + tid];
      sc[ch * CHUNK + tid] = t8;
    }
    __syncthreads();                     // protect pr/buf before next chunk
  }

  // ---- phase 2a: block max ----
  float m = -3.402823466e38f;
  for (int j = tid; j < T_; j += 256) m = fmaxf(m, sc[j]);
#pragma unroll
  for (int o = 16; o > 0; o >>= 1) m = fmaxf(m, __shfl_xor(m, o, 32));
  if (lane == 0) red[wave] = m;
  __syncthreads();
  if (tid == 0) {
    float mm = red[0];
    for (int w = 1; w < 8; ++w) mm = fmaxf(mm, red[w]);
    red[0] = mm;
  }
  __syncthreads();
  m = red[0];
  __syncthreads();

  // ---- phase 2b: exp*mask + block sum, then normalize ----
  float s = 0.f;
  for (int j = tid; j < T_; j += 256) {
    float e = __expf(sc[j] - m) * mask[(size_t)b * T_ + j];
    sc[j] = e;
    s += e;
  }
#pragma unroll
  for (int o = 16; o > 0; o >>= 1) s += __shfl_xor(s, o, 32);
  if (lane == 0) red[wave] = s;
  __syncthreads();
  if (tid == 0) {
    float ss = 0.f;
    for (int w = 0; w < 8; ++w) ss += red[w];
    red[0] = 1.f / ss;
  }
  __syncthreads();
  float inv = red[0];
  for (int j = tid; j < T_; j += 256) {
    float a = sc[j] * inv;
    sc[j] = a;
    if (b == 0) att_out[j] = a;          // atten_vec = attn[0]
  }
  __syncthreads();

  // ---- phase 3: context c[b,:] (direct coalesced loads, 4x unrolled) ----
  int t0 = wave * (T_ / 8);              // 256 t per wave
  float4 a0 = {0,0,0,0}, a1 = {0,0,0,0}, a2 = {0,0,0,0}, a3 = {0,0,0,0};
  for (int i = 0; i < T_ / 8; i += 4) {
    int t = t0 + i;
    const float* p = hv + base + (size_t)t * H_ + 4 * lane;
    float4 v0 = *(const float4*)(p);
    float4 v1 = *(const float4*)(p + H_);
    float4 v2 = *(const float4*)(p + 2 * H_);
    float4 v3 = *(const float4*)(p + 3 * H_);
    float w0 = sc[t], w1 = sc[t + 1], w2 = sc[t + 2], w3 = sc[t + 3];
    a0.x += w0 * v0.x; a0.y += w0 * v0.y; a0.z += w0 * v0.z; a0.w += w0 * v0.w;
    a1.x += w1 * v1.x; a1.y += w1 * v1.y; a1.z += w1 * v1.z; a1.w += w1 * v1.w;
    a2.x += w2 * v2.x; a2.y += w2 * v2.y; a2.z += w2 * v2.z; a2.w += w2 * v2.w;
    a3.x += w3 * v3.x; a3.y += w3 * v3.y; a3.z += w3 * v3.z; a3.w += w3 * v3.w;
  }
  float4 acc;
  acc.x = (a0.x + a1.x) + (a2.x + a3.x);
  acc.y = (a0.y + a1.y) + (a2.y + a3.y);
  acc.z = (a0.z + a1.z) + (a2.z + a3.z);
  acc.w = (a0.w + a1.w) + (a2.w + a3.w);
  *(float4*)(cacc + wave * H_ + 4 * lane) = acc;
  __syncthreads();
  if (tid < H_) {
    float sum = 0.f;
    for (int w = 0; w < 8; ++w) sum += cacc[w * H_ + tid];
    c_out[(size_t)b * H_ + tid] = sum;
  }
}

// ---------------------------------------------------------------------------
// BatchNorm training stats per column of P[B_, H_] (biased var).
// Biases b1/b2 are per-column constants pre-BN -> cancel exactly; skipped.
// ---------------------------------------------------------------------------
__global__ void bn_stats(const float* __restrict__ P, float* __restrict__ stats) {
  __shared__ float s1[256], s2[256];
  int h = blockIdx.x, t = threadIdx.x;
  float a = P[(size_t)(2 * t) * H_ + h];
  float b = P[(size_t)(2 * t + 1) * H_ + h];
  s1[t] = a + b;
  s2[t] = a * a + b * b;
  __syncthreads();
  for (int o = 128; o > 0; o >>= 1) {
    if (t < o) { s1[t] += s1[t + o]; s2[t] += s2[t + o]; }
    __syncthreads();
  }
  if (t == 0) {
    float mean = s1[0] * (1.f / B_);
    float var  = s2[0] * (1.f / B_) - mean * mean;
    stats[h] = mean;
    stats[H_ + h] = var;
  }
}

// ---------------------------------------------------------------------------
// Normalize + ReLU + final GEMV to [B_, V_]
// ---------------------------------------------------------------------------
__global__ void head_final(const float* __restrict__ P,
                           const float* __restrict__ stats,
                           const float* __restrict__ gamma,
                           const float* __restrict__ beta,
                           const float* __restrict__ W3,
                           const float* __restrict__ b3,
                           float* __restrict__ out) {
  __shared__ float r[H_];
  int b = blockIdx.x, t = threadIdx.x;       // 128 threads
  float x = P[(size_t)b * H_ + t];
  float xn = (x - stats[t]) * rsqrtf(stats[H_ + t] + 1e-5f) * gamma[t] + beta[t];
  r[t] = fmaxf(xn, 0.f);
  __syncthreads();
  if (t < V_) {
    float acc = b3[t];
    for (int h = 0; h < H_; ++h) acc += r[h] * W3[t * H_ + h];
    out[(size_t)b * V_ + t] = acc;
  }
}

// ---------------------------------------------------------------------------
extern "C" void kernel_launch(void* const* d_in, const int* in_sizes, int n_in,
                              void* d_out, int out_size, void* d_ws, size_t ws_size,
                              hipStream_t stream) {
  const float* hk    = (const float*)d_in[0];
  const float* hv    = (const float*)d_in[1];
  const float* y1    = (const float*)d_in[2];
  const float* c1    = (const float*)d_in[3];
  const float* sh_1  = (const float*)d_in[4];   // [2,B,H]
  const float* sc_1  = (const float*)d_in[5];   // [2,B,H]
  const float* maskp = (const float*)d_in[6];
  const float* Wih0  = (const float*)d_in[7];
  const float* Whh0  = (const float*)d_in[8];
  const float* bih0  = (const float*)d_in[9];
  const float* bhh0  = (const float*)d_in[10];
  const float* Wih1  = (const float*)d_in[11];
  const float* Whh1  = (const float*)d_in[12];
  const float* bih1  = (const float*)d_in[13];
  const float* bhh1  = (const float*)d_in[14];
  const float* W1    = (const float*)d_in[15];
  const float* W2    = (const float*)d_in[17];
  const float* W3    = (const float*)d_in[19];
  const float* b3    = (const float*)d_in[20];
  const float* gamma = (const float*)d_in[21];
  const float* beta  = (const float*)d_in[22];

  // Output layout: out[B,V], c[B,H], sh0, sh1, sc0, sc1, atten_vec[T]
  float* out   = (float*)d_out;
  float* o_out = out;
  float* o_c   = o_out + B_ * V_;
  float* o_sh0 = o_c   + B_ * H_;
  float* o_sh1 = o_sh0 + B_ * H_;
  float* o_sc0 = o_sh1 + B_ * H_;
  float* o_sc1 = o_sc0 + B_ * H_;
  float* o_att = o_sc1 + B_ * H_;

  // Workspace panels (floats)
  float* ws     = (float*)d_ws;
  float* wA     = ws;                  // up to 512*512
  float* wW     = ws + 262144;         // up to 512*512
  float* wG     = ws + 524288;         // up to 512*512
  float* wStats = ws + 786432;         // 256

  const int TB = 256;

  // ===== LSTM cell 0: gates = [y1|c1|sh_1[0]] (K=512) x [Wih0|Whh0]^T =====
  pack3_kernel<<<(B_ * 512 + TB - 1) / TB, TB, 0, stream>>>(
      y1, E_, c1, H_, sh_1, H_, wA, B_);
  pack2_kernel<<<(G4H * 512 + TB - 1) / TB, TB, 0, stream>>>(
      Wih0, E_ + H_, Whh0, H_, wW, G4H);
  {
    int tiles = (B_ / 16) * (G4H / 16);
    wmma_gemm_nt<<<(tiles + 3) / 4, 128, 0, stream>>>(wA, wW, wG, B_, G4H, 512);
  }
  lstm_elem<<<(B_ * H_ + TB - 1) / TB, TB, 0, stream>>>(
      wG, bih0, bhh0, sc_1, o_sh0, o_sc0);

  // ===== LSTM cell 1: gates = [sh0|sh_1[1]] (K=256) x [Wih1|Whh1]^T =====
  pack2_kernel<<<(B_ * 256 + TB - 1) / TB, TB, 0, stream>>>(
      o_sh0, H_, sh_1 + B_ * H_, H_, wA, B_);
  pack2_kernel<<<(G4H * 256 + TB - 1) / TB, TB, 0, stream>>>(
      Wih1, H_, Whh1, H_, wW, G4H);
  {
    int tiles = (B_ / 16) * (G4H / 16);
    wmma_gemm_nt<<<(tiles + 3) / 4, 128, 0, stream>>>(wA, wW, wG, B_, G4H, 256);
  }
  lstm_elem<<<(B_ * H_ + TB - 1) / TB, TB, 0, stream>>>(
      wG, bih1, bhh1, sc_1 + B_ * H_, o_sh1, o_sc1);

  // ===== Fused attention (bandwidth-dominant part) =====
  attention_kernel<<<B_, 256, 0, stream>>>(hk, hv, o_sh1, maskp, o_c, o_att);

  // ===== Head: P = [sh1|c] x [W1|W2]^T ; BN(train) ; ReLU ; x W3^T + b3 =====
  pack2_kernel<<<(B_ * 256 + TB - 1) / TB, TB, 0, stream>>>(
      o_sh1, H_, o_c, H_, wA, B_);
  pack2_kernel<<<(H_ * 256 + TB - 1) / TB, TB, 0, stream>>>(
      W1, H_, W2, H_, wW, H_);
  {
    int tiles = (B_ / 16) * (H_ / 16);
    wmma_gemm_nt<<<(tiles + 3) / 4, 128, 0, stream>>>(wA, wW, wG, B_, H_, 256);
  }
  bn_stats<<<H_, 256, 0, stream>>>(wG, wStats);
  head_final<<<B_, H_, 0, stream>>>(wG, wStats, gamma, beta, W3, b3, o_out);
}